// APPNP_19413252178060
// MI455X (gfx1250) — compile-verified
//
#include <hip/hip_runtime.h>
#include <hip/hip_bf16.h>
#include <stdint.h>

#define DIM    48
#define KSTEPS 10
#define ALPHA  0.1f
#define WAVES_PER_BLOCK 8

typedef unsigned int u32x4 __attribute__((ext_vector_type(4)));
typedef int          i32x4t __attribute__((ext_vector_type(4)));
typedef int          i32x8t __attribute__((ext_vector_type(8)));

// generic -> LDS(byte offset) conversion for async/TDM descriptors
__device__ __forceinline__ uint32_t lds_offset_u32(const void* p) {
    return (uint32_t)(uintptr_t)(__attribute__((address_space(3))) const void*)p;
}

// -------------------------------------------------------------------------
// Kernel 1: zero an int region (degrees + fill counters).
__global__ void zero_i32_kernel(int* __restrict__ p, int n) {
    int i = blockIdx.x * blockDim.x + threadIdx.x;
    if (i < n) p[i] = 0;
}

// Kernel 2: degree histograms (int atomics, order-independent).
__global__ void degree_kernel(const int* __restrict__ src, const int* __restrict__ dst,
                              int* __restrict__ out_deg, int* __restrict__ in_deg, int e) {
    int i = blockIdx.x * blockDim.x + threadIdx.x;
    if (i < e) {
        atomicAdd(&out_deg[src[i]], 1);
        atomicAdd(&in_deg[dst[i]], 1);
    }
}

// Kernel 3: norms = clip(deg,1)^-0.5
__global__ void norm_kernel(const int* __restrict__ out_deg, const int* __restrict__ in_deg,
                            float* __restrict__ src_norm, float* __restrict__ dst_norm, int n) {
    int i = blockIdx.x * blockDim.x + threadIdx.x;
    if (i < n) {
        src_norm[i] = 1.0f / sqrtf((float)max(out_deg[i], 1));
        dst_norm[i] = 1.0f / sqrtf((float)max(in_deg[i], 1));
    }
}

// Kernels 4a/4b/4c: parallel exclusive scan of in_deg -> row_ptr.
// (three tiny kernels spread across all WGPs instead of one serial block)
__global__ void block_reduce_kernel(const int* __restrict__ deg, int* __restrict__ blk, int n) {
    __shared__ int sm[256];
    const int tid = threadIdx.x;
    const int idx = blockIdx.x * 256 + tid;
    sm[tid] = (idx < n) ? deg[idx] : 0;
    __syncthreads();
    for (int off = 128; off > 0; off >>= 1) {
        if (tid < off) sm[tid] += sm[tid + off];
        __syncthreads();
    }
    if (tid == 0) blk[blockIdx.x] = sm[0];
}

// scans up to 1024 block sums in-place to exclusive offsets (nb = 391 here)
__global__ void scan_blocksums_kernel(int* __restrict__ blk, int nb) {
    __shared__ int sm[1024];
    const int tid = threadIdx.x;
    const int v = (tid < nb) ? blk[tid] : 0;
    sm[tid] = v;
    __syncthreads();
    for (int off = 1; off < 1024; off <<= 1) {
        int t = (tid >= off) ? sm[tid - off] : 0;
        __syncthreads();
        sm[tid] += t;
        __syncthreads();
    }
    if (tid < nb) blk[tid] = sm[tid] - v;   // exclusive
}

__global__ void block_scan_kernel(const int* __restrict__ deg, const int* __restrict__ blk,
                                  int* __restrict__ row_ptr, int n) {
    __shared__ int sm[256];
    const int tid = threadIdx.x;
    const int idx = blockIdx.x * 256 + tid;
    const int v = (idx < n) ? deg[idx] : 0;
    sm[tid] = v;
    __syncthreads();
    for (int off = 1; off < 256; off <<= 1) {
        int t = (tid >= off) ? sm[tid - off] : 0;
        __syncthreads();
        sm[tid] += t;
        __syncthreads();
    }
    if (idx < n) row_ptr[idx + 1] = blk[blockIdx.x] + sm[tid];
    if (idx == 0) row_ptr[0] = 0;
}

// Kernel 5: CSR scatter; fold src_norm into the edge record.
__global__ void scatter_kernel(const int* __restrict__ src, const int* __restrict__ dst,
                               const float* __restrict__ src_norm,
                               const int* __restrict__ row_ptr, int* __restrict__ fill,
                               int2* __restrict__ edges, int e) {
    int i = blockIdx.x * blockDim.x + threadIdx.x;
    if (i < e) {
        const int d = dst[i];
        const int s = src[i];
        const int pos = atomicAdd(&fill[d], 1);
        edges[row_ptr[d] + pos] = make_int2(s, __float_as_int(src_norm[s]));
    }
}

// Kernel 6: one propagation step. One wave32 per node; lanes 0..23 own a
// float2 slice of the 48-wide row (one global_load_b64 covers the 192B row,
// L2-resident: whole working set ~57MB << 192MB L2). Edge (src,weight) pairs
// are DMA'd into per-wave LDS staging via GLOBAL_LOAD_ASYNC_TO_LDS_B64
// (ASYNCcnt), then read with uniform-address ds_load_b64 broadcasts.
// s_wait_dscnt before each async issue closes the WAR hazard between the
// previous chunk's in-flight DS reads and the async engine's LDS write
// (ASYNCcnt and DScnt are independent per the ISA).
__global__ __launch_bounds__(256) void prop_kernel(
    const float* __restrict__ feat, const float* __restrict__ feat0,
    const int2* __restrict__ edges, const int* __restrict__ row_ptr,
    const float* __restrict__ dst_norm, float* __restrict__ out, int n) {

    __shared__ int2 stage[WAVES_PER_BLOCK][32];   // 256B per wave

    const int lane = threadIdx.x & 31;
    const int wib  = threadIdx.x >> 5;
    const int node = (int)((blockIdx.x * blockDim.x + threadIdx.x) >> 5);
    if (node >= n) return;

    int2* sbuf = &stage[wib][0];
    const uint32_t lds_base = lds_offset_u32(sbuf);

    const int beg = row_ptr[node];
    const int end = row_ptr[node + 1];
    const bool active = lane < 24;      // 24 lanes x float2 = 48 features
    const int fo = lane * 2;

    float ax = 0.0f, ay = 0.0f;
    for (int c = beg; c < end; c += 32) {
        const int cnt = min(32, end - c);
        if (lane == 0 && c + 32 < end)
            __builtin_prefetch(edges + c + 32, 0, 1);   // global_prefetch_b8
        asm volatile("s_wait_dscnt 0x0" ::: "memory");  // prior DS reads done
        if (lane < cnt) {
            asm volatile("global_load_async_to_lds_b64 %0, %1, off"
                         :: "v"(lds_base + (uint32_t)lane * 8u),
                            "v"((uint64_t)(uintptr_t)(edges + c + lane))
                         : "memory");
        }
        asm volatile("s_wait_asynccnt 0x0" ::: "memory");
        #pragma unroll 4
        for (int t = 0; t < cnt; ++t) {
            const int2  pr = sbuf[t];                   // ds_load_b64 broadcast
            const int   s  = pr.x;
            const float w  = __int_as_float(pr.y);
            if (active) {
                const float2 fr = *(const float2*)(&feat[s * DIM + fo]);
                ax = fmaf(fr.x, w, ax);
                ay = fmaf(fr.y, w, ay);
            }
        }
    }
    if (active) {
        const float dn = dst_norm[node] * (1.0f - ALPHA);
        const float2 f0 = *(const float2*)(&feat0[node * DIM + fo]);
        float2 r;
        r.x = fmaf(dn, ax, ALPHA * f0.x);
        r.y = fmaf(dn, ay, ALPHA * f0.y);
        *(float2*)(&out[node * DIM + fo]) = r;
    }
}

// -------------------------------------------------------------------------
// TDM reference (compiled, intentionally not launched): gather-mode Tensor-DMA
// descriptor gathering 8 rows (32-bit indices) of the 48-wide f32 feature
// tensor into LDS via tensor_load_to_lds / TENSORcnt. Kept as
// encoding documentation; not used in the hot path because 1.5KB gathers per
// descriptor (+ per-batch descriptor assembly and s_wait_tensorcnt) lose to
// direct cacheline-sized L2 gathers for this working set.
__global__ void tdm_gather_probe_kernel(const float* __restrict__ feat,
                                        const int* __restrict__ rows,
                                        float* __restrict__ out, int ntensor) {
    __shared__ float tile[8 * DIM];
    if (threadIdx.x == 0) {
        const uint64_t ga  = (uint64_t)(uintptr_t)feat;
        const uint32_t lds = lds_offset_u32(&tile[0]);
        u32x4 g0;
        g0.x = 1u | (1u << 30) | (1u << 31);          // count=1, 32b indices, gather_mode
        g0.y = lds;                                    // lds_addr  [63:32]
        g0.z = (uint32_t)ga;                           // global_addr[31:0]
        g0.w = ((uint32_t)(ga >> 32) & 0x01FFFFFFu)    // global_addr[56:32]
             | (2u << 30);                             // type=2
        i32x8t g1;
        g1[0] = (2 << 16);                             // data_size=4B, wg_mask=0
        g1[1] = (DIM << 16);                           // tensor_dim0 lo
        g1[2] = ((ntensor & 0xFFFF) << 16);            // tensor_dim1 lo
        g1[3] = ((ntensor >> 16) & 0xFFFF)             // tensor_dim1 hi
              | (DIM << 16);                           // tile_dim0 = 48
        g1[4] = 8;                                     // tile_dim1 = #valid indices
        g1[5] = DIM;                                   // tensor_dim0_stride
        g1[6] = 0;
        g1[7] = 0;
        i32x4t g2, g3;
        g2[0] = rows[0]; g2[1] = rows[1]; g2[2] = rows[2]; g2[3] = rows[3];
        g3[0] = rows[4]; g3[1] = rows[5]; g3[2] = rows[6]; g3[3] = rows[7];
        asm volatile("tensor_load_to_lds %0, %1, %2, %3"
                     :: "s"(g0), "s"(g1), "s"(g2), "s"(g3) : "memory");
    }
    __builtin_amdgcn_s_wait_tensorcnt(0);
    __syncthreads();
    if ((int)threadIdx.x < 8 * DIM) out[threadIdx.x] = tile[threadIdx.x];
}

// -------------------------------------------------------------------------
extern "C" void kernel_launch(void* const* d_in, const int* in_sizes, int n_in,
                              void* d_out, int out_size, void* d_ws, size_t ws_size,
                              hipStream_t stream) {
    const float* features = (const float*)d_in[0];
    const int*   src      = (const int*)d_in[1];
    const int*   dst      = (const int*)d_in[2];
    const int n = in_sizes[0] / DIM;     // 100000
    const int e = in_sizes[1];           // 1600000
    const int nb = (n + 255) / 256;      // 391 scan blocks (fits one 1024-block)

    int*   out_deg  = (int*)d_ws;                 // n
    int*   in_deg   = out_deg + n;                // n
    int*   fill     = in_deg + n;                 // n (zeroed together: 3n)
    float* src_norm = (float*)(fill + n);         // n
    float* dst_norm = src_norm + n;               // n
    int*   row_ptr  = (int*)(dst_norm + n);       // n+1
    int*   blk      = row_ptr + n + 1;            // nb
    uintptr_t p = (uintptr_t)(blk + nb);
    p = (p + 15) & ~(uintptr_t)15;
    int2*  edges = (int2*)p;                      // e pairs
    float* bufA  = (float*)(edges + e);           // n*DIM floats

    const int T = 256;
    zero_i32_kernel<<<(3 * n + T - 1) / T, T, 0, stream>>>(out_deg, 3 * n);
    degree_kernel<<<(e + T - 1) / T, T, 0, stream>>>(src, dst, out_deg, in_deg, e);
    norm_kernel<<<(n + T - 1) / T, T, 0, stream>>>(out_deg, in_deg, src_norm, dst_norm, n);
    block_reduce_kernel<<<nb, 256, 0, stream>>>(in_deg, blk, n);
    scan_blocksums_kernel<<<1, 1024, 0, stream>>>(blk, nb);
    block_scan_kernel<<<nb, 256, 0, stream>>>(in_deg, blk, row_ptr, n);
    scatter_kernel<<<(e + T - 1) / T, T, 0, stream>>>(src, dst, src_norm, row_ptr, fill, edges, e);

    const int blocks = (n + WAVES_PER_BLOCK - 1) / WAVES_PER_BLOCK;
    const float* cur = features;
    for (int k = 0; k < KSTEPS; ++k) {
        // K=10 (even): odd steps write d_out -> step 10 lands in d_out
        float* nxt = (k & 1) ? (float*)d_out : bufA;
        prop_kernel<<<blocks, T, 0, stream>>>(cur, features, edges, row_ptr, dst_norm, nxt, n);
        cur = nxt;
    }
}